// SCL_Dec_24077586662024
// MI455X (gfx1250) — compile-verified
//
#include <hip/hip_runtime.h>

#define NN      128
#define STAGESN 7
#define L2N     16
#define LISTN   8
#define THREADS 256
#define KFROZEN 64
#define LLRMAXF 30.0f

// All decoder state for one codeword lives in LDS (~115 KB, fits 320 KB WGP LDS).
struct __align__(16) SMem {
    float llr[L2N][STAGESN][NN];   // stages 0..6 (stage 7 == shared channel LLRs)
    float uh [L2N][STAGESN][NN];   // stages 0..6 (stage-7 uhat is never read)
    float ch[NN];                  // raw inputs; negate on read (llr_ch = -inputs)
    float pm[L2N];
    float pmN[L2N];
    int   ptr[L2N];
    int   ptrN[L2N];
};

__device__ __forceinline__ float clampf(float x) {
    return fminf(fmaxf(x, -LLRMAXF), LLRMAXF);
}
__device__ __forceinline__ float softplusf(float x) {
    // jax.nn.softplus(x) = max(x,0) + log1p(exp(-|x|))
    return fmaxf(x, 0.0f) + log1pf(expf(-fabsf(x)));
}
__device__ __forceinline__ float ldllr(SMem* sm, int p, int s, int i) {
    return (s == STAGESN) ? -sm->ch[i] : sm->llr[p][s][i];
}
__device__ __forceinline__ void wbar() { __builtin_amdgcn_wave_barrier(); }

// ---------------- Phase B helpers: all 16 rows, block-wide ----------------

// f: llr[s-1][start+j] = sign(x)*sign(y)*min(|x|,|y|) with clipped inputs
__device__ __forceinline__ void f_stage(SMem* sm, int s, int start) {
    const int h = 1 << (s - 1);
    for (int t = threadIdx.x; t < L2N * h; t += THREADS) {
        const int p = t >> (s - 1);
        const int j = t & (h - 1);
        float x = clampf(ldllr(sm, p, s, start + j));
        float y = clampf(ldllr(sm, p, s, start + h + j));
        float m = fminf(fabsf(x), fabsf(y));
        unsigned sgn = (__float_as_uint(x) ^ __float_as_uint(y)) & 0x80000000u;
        sm->llr[p][s - 1][start + j] = __uint_as_float(__float_as_uint(m) | sgn);
    }
}

// g: llr[s-1][start+h+j] = (1-2u)*x + y   (no clipping, matches reference)
__device__ __forceinline__ void g_stage(SMem* sm, int s, int start) {
    const int h = 1 << (s - 1);
    for (int t = threadIdx.x; t < L2N * h; t += THREADS) {
        const int p = t >> (s - 1);
        const int j = t & (h - 1);
        float x = ldllr(sm, p, s, start + j);
        float y = ldllr(sm, p, s, start + h + j);
        float u = sm->uh[p][s - 1][start + j];
        sm->llr[p][s - 1][start + h + j] = (1.0f - 2.0f * u) * x + y;
    }
}

// combine: uh[s][start+j] = ul XOR ur (as floats), uh[s][start+h+j] = ur
__device__ __forceinline__ void combine(SMem* sm, int s, int start) {
    const int h = 1 << (s - 1);
    for (int t = threadIdx.x; t < L2N * h; t += THREADS) {
        const int p = t >> (s - 1);
        const int j = t & (h - 1);
        float ul = sm->uh[p][s - 1][start + j];
        float ur = sm->uh[p][s - 1][start + h + j];
        sm->uh[p][s][start + j]     = ul + ur - 2.0f * ul * ur;
        sm->uh[p][s][start + h + j] = ur;
    }
}

// info leaf: pm update, stable 16-way sort, survivor copies (right half only)
__device__ void leaf_info(SMem* sm, int i) {
    const int tid = threadIdx.x;
    if (tid < L2N) {
        int row = sm->ptr[tid];
        float u;
        if (tid >= LISTN) { sm->uh[row][0][i] = 1.0f; u = 1.0f; }
        else              { u = sm->uh[row][0][i]; }
        float l = clampf(sm->llr[row][0][i]);
        sm->pm[tid] += softplusf(-(1.0f - 2.0f * u) * l);
    }
    __syncthreads();

    // stable rank sort of 16 (pm, ptr) pairs — matches jnp.argsort stability
    if (tid < L2N) {
        float mypm = sm->pm[tid];
        int rank = 0;
        #pragma unroll
        for (int j = 0; j < L2N; ++j) {
            float pj = sm->pm[j];
            rank += (pj < mypm) || ((pj == mypm) && (j < tid));
        }
        sm->pmN[rank]  = mypm;
        sm->ptrN[rank] = sm->ptr[tid];
    }
    __syncthreads();
    if (tid < L2N) {
        sm->pm[tid]  = sm->pmN[tid & (LISTN - 1)];   // pm[8:] = pm[:8]
        sm->ptr[tid] = sm->ptrN[tid];
    }
    // survivor copies: rows ptrN[8+j] <- ptrN[j], both arrays, stages 0..6,
    // columns 64..127 only (left half is provably dead after the root g).
    const int PER = STAGESN * (KFROZEN / 4);   // 112 float4 per array per pair
    const int TOT = LISTN * PER * 2;           // 1792
    for (int t = tid; t < TOT; t += THREADS) {
        int pair = t / (PER * 2);
        int r    = t - pair * (PER * 2);
        int arr  = r / PER;
        int q    = r - arr * PER;
        int s    = q >> 4;                     // stage
        int o    = q & 15;                     // float4 within the 64-col half
        int src  = sm->ptrN[pair];
        int dst  = sm->ptrN[LISTN + pair];
        const float4* s4 = (const float4*)(arr ? &sm->uh[src][s][KFROZEN]
                                               : &sm->llr[src][s][KFROZEN]);
        float4*       d4 = (float4*)      (arr ? &sm->uh[dst][s][KFROZEN]
                                               : &sm->llr[dst][s][KFROZEN]);
        d4[o] = s4[o];
    }
    __syncthreads();
}

// ---------------- Phase A: frozen half, row 0 only, wave 0, no barriers ----
// All partial sums are provably zero in the frozen half, so g degenerates to
// x+y and combines are skipped (outputs stay at their reference init of 0).
__device__ void phaseA(SMem* sm, int lane) {
    auto f0 = [&](int s, int start) {
        int h = 1 << (s - 1);
        for (int j = lane; j < h; j += 32) {
            float x = clampf(ldllr(sm, 0, s, start + j));
            float y = clampf(ldllr(sm, 0, s, start + h + j));
            float m = fminf(fabsf(x), fabsf(y));
            unsigned sgn = (__float_as_uint(x) ^ __float_as_uint(y)) & 0x80000000u;
            sm->llr[0][s - 1][start + j] = __uint_as_float(__float_as_uint(m) | sgn);
        }
        wbar();
    };
    auto g0 = [&](int s, int start) {       // u == 0  ->  x + y
        int h = 1 << (s - 1);
        for (int j = lane; j < h; j += 32) {
            float x = ldllr(sm, 0, s, start + j);
            float y = ldllr(sm, 0, s, start + h + j);
            sm->llr[0][s - 1][start + h + j] = x + y;
        }
        wbar();
    };
    float S = 0.0f;                          // shared frozen-metric increment
    for (int i = 0; i < KFROZEN; ++i) {
        if (i == 0) {
            for (int s = STAGESN; s >= 1; --s) f0(s, 0);
        } else {
            int t = __ffs(i) - 1;            // ctz(i) <= 5 here
            g0(t + 1, i - (1 << t));
            for (int s = t; s >= 1; --s) f0(s, i);
        }
        if (lane == 0) S += softplusf(-clampf(sm->llr[0][0][i]));
        wbar();
    }
    g0(STAGESN, 0);                          // root g: llr[6][64..127] (u==0)
    if (lane == 0) sm->pmN[0] = S;           // stash S for the transition
    wbar();
}

extern "C" __global__ void __launch_bounds__(THREADS)
scl_decode_kernel(const float* __restrict__ in, float* __restrict__ out) {
    extern __shared__ char smem_raw[];
    SMem* sm = (SMem*)smem_raw;
    const int b   = blockIdx.x;
    const int tid = threadIdx.x;

    // CDNA5 async DMA: channel LLRs (512 B) global -> LDS, one b128 per lane
    // of wave 0, tracked by ASYNCcnt; overlapped with the zero-init below.
    if (tid < 32) {
        unsigned lds_b = (unsigned)(size_t)(&sm->ch[tid * 4]);
        const float* g = in + (size_t)b * NN + tid * 4;
        asm volatile("global_load_async_to_lds_b128 %0, %1, off"
                     :: "v"(lds_b), "v"(g) : "memory");
    }

    // zero only what is read before written: uh[*][*][64..128)
    {
        float4 z = make_float4(0.f, 0.f, 0.f, 0.f);
        const int TOT4 = L2N * STAGESN * (KFROZEN / 4);          // 1792
        for (int t = tid; t < TOT4; t += THREADS) {
            int row = t / (STAGESN * 16);
            int r   = t - row * (STAGESN * 16);
            int s   = r >> 4;
            int o   = r & 15;
            ((float4*)&sm->uh[row][s][KFROZEN])[o] = z;
        }
    }
    if (tid < 32) {
        asm volatile("s_wait_asynccnt 0" ::: "memory");
    }
    __syncthreads();

    // ---- Phase A: frozen half on row 0, wave 0 only, barrier-free ----
    if (tid < 32) phaseA(sm, tid);
    __syncthreads();

    // ---- Transition: replicate live row-0 state, init pm/ptr ----
    {
        float S = sm->pmN[0];
        for (int t = tid; t < 15 * (KFROZEN / 4); t += THREADS) {  // 240 float4
            int row = 1 + t / (KFROZEN / 4);
            int o   = t & 15;
            ((float4*)&sm->llr[row][STAGESN - 1][KFROZEN])[o] =
                ((float4*)&sm->llr[0][STAGESN - 1][KFROZEN])[o];
        }
        if (tid < L2N) {
            sm->pm[tid]  = ((tid == 0 || tid == LISTN) ? 0.0f : LLRMAXF) + S;
            sm->ptr[tid] = tid;
        }
    }
    __syncthreads();

    // ---- Phase B: info half, full 16-row list decoding ----
    for (int i = KFROZEN; i < NN; ++i) {
        if (i == KFROZEN) {
            // descent into the right half (root g already done in phase A)
            for (int s = STAGESN - 1; s >= 1; --s) { f_stage(sm, s, KFROZEN); __syncthreads(); }
        } else {
            const int t = __ffs(i) - 1;                          // ctz(i) <= 5
            for (int s = 1; s <= t; ++s) { combine(sm, s, i - (1 << s)); __syncthreads(); }
            g_stage(sm, t + 1, i - (1 << t)); __syncthreads();
            for (int s = t; s >= 1; --s) { f_stage(sm, s, i); __syncthreads(); }
        }
        leaf_info(sm, i);
    }

    // best path after the final info-leaf sort is slot 0 (stable argmin)
    const int best = sm->ptr[0];
    if (tid < (NN - KFROZEN)) {
        out[(size_t)b * (NN - KFROZEN) + tid] = sm->uh[best][0][KFROZEN + tid];
    }
}

extern "C" void kernel_launch(void* const* d_in, const int* in_sizes, int n_in,
                              void* d_out, int out_size, void* d_ws, size_t ws_size,
                              hipStream_t stream) {
    (void)n_in; (void)d_ws; (void)ws_size; (void)out_size;
    const float* in = (const float*)d_in[0];
    float* out = (float*)d_out;
    const int batch = in_sizes[0] / NN;                          // 2048
    const size_t shmem = sizeof(SMem);                           // ~115 KB
    static_assert(sizeof(SMem) <= 320 * 1024, "LDS budget");
    (void)hipFuncSetAttribute((const void*)scl_decode_kernel,
                              hipFuncAttributeMaxDynamicSharedMemorySize,
                              (int)shmem);
    scl_decode_kernel<<<dim3(batch), dim3(THREADS), shmem, stream>>>(in, out);
}